// DynamicRouting2d_23132693856552
// MI455X (gfx1250) — compile-verified
//
#include <hip/hip_runtime.h>
#include <math.h>

// CDNA5 (gfx1250) fused capsule dynamic-routing kernel.
// One workgroup = (batch, tile of 16 spatial positions). pose_out is never
// materialized to global memory: it is recomputed per routing iteration with
// V_WMMA_F32_16X16X4_F32 (fp32 A/B, K=8 via two k-steps), and routing state
// lives in the WMMA accumulator register layout.

typedef __attribute__((ext_vector_type(2))) float v2f;
typedef __attribute__((ext_vector_type(8))) float v8f;

#define A_      32
#define B_      32
#define C_      8
#define D_      16
#define KK_     9
#define KKA_    288      // KK * A
#define H_      14
#define L_      196      // 14*14
#define BD_     512      // B * D
#define POS_    16       // positions per workgroup (WMMA N)
#define THREADS_ 256     // 8 wave32
#define ITERS_  3

__global__ __launch_bounds__(THREADS_, 1)
void caps_route_kernel(const float* __restrict__ pose,
                       const float* __restrict__ W,
                       float* __restrict__ out)
{
    // Patch tensor: Xs[k][c][pos], k = kk*32 + a, value = pose[b, a*8+c, y+dy-1, x+dx-1]
    __shared__ float Xs[KKA_ * C_ * POS_];    // 147456 B
    __shared__ float lgbuf[2][B_ * POS_];     // 4096 B, double-buffered by k parity

    const int tid   = threadIdx.x;
    const int l0    = blockIdx.x * POS_;
    const int batch = blockIdx.y;

    // ---- Phase 0: cooperative patch gather (zero-padded conv patches) ----
    for (int idx = tid; idx < KKA_ * C_ * POS_; idx += THREADS_) {
        const int pos = idx & (POS_ - 1);
        const int rem = idx >> 4;
        const int c   = rem & (C_ - 1);
        const int k   = rem >> 3;          // 0..287
        const int kk  = k >> 5;            // kernel tap 0..8
        const int a   = k & 31;            // input capsule
        const int l   = l0 + pos;
        float val = 0.0f;
        if (l < L_) {
            const int y  = l / H_;
            const int x  = l - y * H_;
            const int py = y + (kk / 3) - 1;
            const int px = x + (kk % 3) - 1;
            if (py >= 0 && py < H_ && px >= 0 && px < H_) {
                val = pose[((batch * (A_ * C_) + a * C_ + c) * H_ + py) * H_ + px];
            }
        }
        Xs[idx] = val;
    }
    __syncthreads();

    const int lane = tid & 31;
    const int wave = tid >> 5;       // 0..7, owns output capsules 4*wave..4*wave+3
    const int pos  = lane & 15;      // WMMA N / column
    const int hi   = lane >> 4;      // half-wave select
    const int cb   = hi ? 2 : 0;     // K-subrow pair per ISA A/B operand layout

    // Accumulated v over previous routing iterations (logits = <Vacc, pose_out>).
    // Layout identical to WMMA C/D: vacc[tt][r] = (b_out=4*wave+tt, d=r+8*hi, pos).
    float vacc[4][8];
#pragma unroll
    for (int tt = 0; tt < 4; ++tt)
#pragma unroll
        for (int r = 0; r < 8; ++r) vacc[tt][r] = 0.0f;

    const v8f czero = {0.f, 0.f, 0.f, 0.f, 0.f, 0.f, 0.f, 0.f};

    for (int it = 0; it < ITERS_; ++it) {
        float sacc[4][8];
#pragma unroll
        for (int tt = 0; tt < 4; ++tt)
#pragma unroll
            for (int r = 0; r < 8; ++r) sacc[tt][r] = 0.0f;

        for (int k = 0; k < KKA_; ++k) {
            // Prefetch next k's W slab (speculative; OOB at k==287 is dropped).
            __builtin_prefetch(W + (size_t)(k + 1) * BD_ * C_, 0, 0);

            // B operand (4x16 per k-step), shared by all M-tiles:
            // vgpr0 = K-rows {cb, cb} , vgpr1 = K-rows {cb+1}; k-step 1 at c+4.
            const float* xk = Xs + k * (C_ * POS_);
            v2f b0, b1;
            b0.x = xk[(cb + 0) * POS_ + pos];
            b0.y = xk[(cb + 1) * POS_ + pos];
            b1.x = xk[(cb + 4) * POS_ + pos];
            b1.y = xk[(cb + 5) * POS_ + pos];

            v8f   p[4];
            float lgt[4];
#pragma unroll
            for (int tt = 0; tt < 4; ++tt) {
                const int row = ((wave * 4 + tt) * D_) + pos;   // A-row = M = p-index
                const float* wp = W + ((size_t)k * BD_ + row) * C_ + cb;
                v2f a0, a1;
                a0.x = wp[0]; a0.y = wp[1];     // K = cb, cb+1
                a1.x = wp[4]; a1.y = wp[5];     // K = cb+4, cb+5
                v8f acc = __builtin_amdgcn_wmma_f32_16x16x4_f32(
                    false, a0, false, b0, (short)0, czero, false, false);
                acc = __builtin_amdgcn_wmma_f32_16x16x4_f32(
                    false, a1, false, b1, (short)0, acc, false, false);
                p[tt] = acc;

                // logits[k, b_out, pos] = sum_d Vacc * pose_out (8 in-lane + partner)
                float lg = 0.0f;
#pragma unroll
                for (int r = 0; r < 8; ++r) lg += vacc[tt][r] * acc[r];
                lg += __shfl_xor(lg, 16, 32);
                lgt[tt] = lg;
            }

            // Publish logits for cross-capsule softmax (one writer half-wave).
            float* lb = lgbuf[k & 1];
            if (!hi) {
#pragma unroll
                for (int tt = 0; tt < 4; ++tt)
                    lb[(wave * 4 + tt) * POS_ + pos] = lgt[tt];
            }
            __syncthreads();

            // Softmax over the 32 output capsules for this (k, pos).
            float mx = -INFINITY;
#pragma unroll
            for (int j = 0; j < B_; ++j) mx = fmaxf(mx, lb[j * POS_ + pos]);
            float den = 0.0f;
#pragma unroll
            for (int j = 0; j < B_; ++j) den += __expf(lb[j * POS_ + pos] - mx);
            const float inv = 1.0f / den;

#pragma unroll
            for (int tt = 0; tt < 4; ++tt) {
                const float cij = __expf(lgt[tt] - mx) * inv;
#pragma unroll
                for (int r = 0; r < 8; ++r) sacc[tt][r] += cij * p[tt][r];
            }
        }

        // ---- squash(s) per (b_out, pos); update Vacc or emit output ----
#pragma unroll
        for (int tt = 0; tt < 4; ++tt) {
            float msq = 0.0f;
#pragma unroll
            for (int r = 0; r < 8; ++r) msq += sacc[tt][r] * sacc[tt][r];
            msq += __shfl_xor(msq, 16, 32);
            const float mag   = sqrtf(msq);
            const float scale = msq / ((1.0f + msq) * fmaxf(mag, 1e-30f));
            if (it + 1 < ITERS_) {
#pragma unroll
                for (int r = 0; r < 8; ++r) vacc[tt][r] += sacc[tt][r] * scale;
            } else {
                const int l = l0 + pos;
                if (l < L_) {
                    const int ch_base = (wave * 4 + tt) * D_ + hi * 8;
#pragma unroll
                    for (int r = 0; r < 8; ++r) {
                        out[((size_t)batch * BD_ + ch_base + r) * L_ + l] =
                            sacc[tt][r] * scale;
                    }
                }
            }
        }
        __syncthreads();   // protect lgbuf reuse across iterations
    }
}

extern "C" void kernel_launch(void* const* d_in, const int* in_sizes, int n_in,
                              void* d_out, int out_size, void* d_ws, size_t ws_size,
                              hipStream_t stream) {
    (void)in_sizes; (void)n_in; (void)d_ws; (void)ws_size; (void)out_size;
    const float* pose = (const float*)d_in[0];   // (4, 256, 14, 14) fp32
    const float* W    = (const float*)d_in[1];   // (288, 512, 8) fp32
    float*       out  = (float*)d_out;           // (4, 512, 14, 14) fp32

    dim3 grid((L_ + POS_ - 1) / POS_, 4);        // 13 position tiles x 4 batches
    dim3 block(THREADS_);
    caps_route_kernel<<<grid, block, 0, stream>>>(pose, W, out);
}